// GNNRefiner_30202210025926
// MI455X (gfx1250) — compile-verified
//
#include <hip/hip_runtime.h>
#include <hip/hip_bf16.h>
#include <stdint.h>

typedef __attribute__((ext_vector_type(2))) float v2f;
typedef __attribute__((ext_vector_type(8))) float v8f;

// ---------------------------------------------------------------------------
// 1) deg[n] = 1.0f  (self-loop contributes 1 to every node's in-degree)
// ---------------------------------------------------------------------------
__global__ void gcn_init_deg(float* __restrict__ deg, int n) {
    int i = blockIdx.x * blockDim.x + threadIdx.x;
    if (i < n) deg[i] = 1.0f;
}

// ---------------------------------------------------------------------------
// 2) deg[dst[e]] += 1  for the 5M real edges (non-returning f32 atomics)
// ---------------------------------------------------------------------------
__global__ void gcn_deg_count(const long long* __restrict__ edges,
                              float* __restrict__ deg, long long E) {
    long long e = (long long)blockIdx.x * blockDim.x + threadIdx.x;
    if (e < E) {
        int d = (int)edges[E + e];   // dst half of [2,E] int64 edge_index
        atomicAdd(&deg[d], 1.0f);
    }
}

// ---------------------------------------------------------------------------
// 3) dinv = rsqrt(deg)  (deg >= 1 always); seed s1 with the self-loop term:
//    s1[n] = dinv[n]^2 * x[n]
// ---------------------------------------------------------------------------
__global__ void gcn_dinv_seed(float* __restrict__ deg_dinv,  // in: deg, out: dinv
                              const float* __restrict__ x,
                              float* __restrict__ s1, int n) {
    int i = blockIdx.x * blockDim.x + threadIdx.x;
    if (i < n) {
        float d = __frsqrt_rn(deg_dinv[i]);
        deg_dinv[i] = d;
        s1[i] = d * d * x[i];
    }
}

// ---------------------------------------------------------------------------
// 4) layer-1 scalar scatter: s1[dst] += dinv[src]*dinv[dst]*x[src]
// ---------------------------------------------------------------------------
__global__ void gcn_scatter1(const long long* __restrict__ edges,
                             const float* __restrict__ dinv,
                             const float* __restrict__ x,
                             float* __restrict__ s1, long long E) {
    long long e = (long long)blockIdx.x * blockDim.x + threadIdx.x;
    if (e < E) {
        int s = (int)edges[e];
        int d = (int)edges[E + e];
        float nrm = dinv[s] * dinv[d];
        atomicAdd(&s1[d], nrm * x[s]);
    }
}

// ---------------------------------------------------------------------------
// 5) Per-node fused middle stage via WMMA.
//    out1[n,c] = relu(s1[n]*W1[c] + b1[c]);  h2[n] = sum_c out1[n,c]*W2[c]
//    One wave32 = one tile of 16 nodes. Channel dim mapped to K; 4 chained
//    v_wmma_f32_16x16x4_f32 accumulate K=16. B = W2 broadcast across N cols.
//    A layout (16x4 f32): lane = half*16 + m, VGPR i -> A[m, half*2 + i].
//    D layout: VGPR j, lane-half h -> (M = j + 8h, N = lane&15).
//    Also seeds h2acc with the layer-2 self-loop term dinv^2 * h2.
// ---------------------------------------------------------------------------
__global__ void gcn_node_wmma(const float* __restrict__ s1,
                              const float* __restrict__ dinv,
                              const float* __restrict__ W1,
                              const float* __restrict__ b1,
                              const float* __restrict__ W2,
                              float* __restrict__ h2,
                              float* __restrict__ h2acc,
                              int n_nodes, int n_tiles) {
    int gtid = blockIdx.x * blockDim.x + threadIdx.x;
    int tile = gtid >> 5;            // wave-uniform
    int lane = gtid & 31;
    if (tile >= n_tiles) return;     // whole-wave exit; EXEC stays all-1s below

    int m    = lane & 15;
    int half = lane >> 4;
    int node = tile * 16 + m;
    float s  = (node < n_nodes) ? s1[node] : 0.0f;

    v8f acc = {};
#pragma unroll
    for (int q = 0; q < 4; ++q) {
        int c0 = q * 4 + half * 2;          // this lane's two K slots in chunk q
        float r0 = fmaf(s, W1[c0],     b1[c0]);
        float r1 = fmaf(s, W1[c0 + 1], b1[c0 + 1]);
        v2f a, b;
        a.x = r0 > 0.0f ? r0 : 0.0f;        // relu
        a.y = r1 > 0.0f ? r1 : 0.0f;
        b.x = W2[c0];                        // broadcast over all 16 N-columns
        b.y = W2[c0 + 1];
        acc = __builtin_amdgcn_wmma_f32_16x16x4_f32(
            /*neg_a=*/false, a, /*neg_b=*/false, b,
            /*c_mod=*/(short)0, acc, /*reuse_a=*/false, /*reuse_b=*/false);
    }

    // Every N-column of D holds the same dot product; lane 0 / lane 16 write
    // rows M=0..7 / M=8..15 of the tile.
    if (m == 0) {
#pragma unroll
        for (int j = 0; j < 8; ++j) {
            int out_node = tile * 16 + half * 8 + j;
            if (out_node < n_nodes) {
                float v = acc[j];
                h2[out_node] = v;
                float di = dinv[out_node];
                h2acc[out_node] = di * di * v;   // layer-2 self-loop seed
            }
        }
    }
}

// ---------------------------------------------------------------------------
// 6) layer-2 scalar scatter: h2acc[dst] += dinv[src]*dinv[dst]*h2[src]
// ---------------------------------------------------------------------------
__global__ void gcn_scatter2(const long long* __restrict__ edges,
                             const float* __restrict__ dinv,
                             const float* __restrict__ h2,
                             float* __restrict__ h2acc, long long E) {
    long long e = (long long)blockIdx.x * blockDim.x + threadIdx.x;
    if (e < E) {
        int s = (int)edges[e];
        int d = (int)edges[E + e];
        float nrm = dinv[s] * dinv[d];
        atomicAdd(&h2acc[d], nrm * h2[s]);
    }
}

// ---------------------------------------------------------------------------
// 7) out[n] = sigmoid(h2acc[n] + b2)
// ---------------------------------------------------------------------------
__global__ void gcn_final(const float* __restrict__ h2acc,
                          const float* __restrict__ b2,
                          float* __restrict__ out, int n) {
    int i = blockIdx.x * blockDim.x + threadIdx.x;
    if (i < n) {
        float z = h2acc[i] + b2[0];
        out[i] = 1.0f / (1.0f + __expf(-z));
    }
}

extern "C" void kernel_launch(void* const* d_in, const int* in_sizes, int n_in,
                              void* d_out, int out_size, void* d_ws, size_t ws_size,
                              hipStream_t stream) {
    const float*     x     = (const float*)d_in[0];
    const long long* edges = (const long long*)d_in[1];   // int64 [2, E] flat
    const float*     W1    = (const float*)d_in[2];       // 16
    const float*     b1    = (const float*)d_in[3];       // 16
    const float*     W2    = (const float*)d_in[4];       // 16
    const float*     b2    = (const float*)d_in[5];       // 1
    float*           out   = (float*)d_out;

    const int       N = in_sizes[0];
    const long long E = (long long)in_sizes[1] / 2;

    // Workspace: 4 float arrays of N elements (deg/dinv, s1, h2, h2acc)
    float* deg_dinv = (float*)d_ws;
    float* s1       = deg_dinv + N;
    float* h2       = s1 + N;
    float* h2acc    = h2 + N;

    const int TB = 256;
    const int nblk_n = (N + TB - 1) / TB;
    const int nblk_e = (int)((E + TB - 1) / TB);

    gcn_init_deg<<<nblk_n, TB, 0, stream>>>(deg_dinv, N);
    gcn_deg_count<<<nblk_e, TB, 0, stream>>>(edges, deg_dinv, E);
    gcn_dinv_seed<<<nblk_n, TB, 0, stream>>>(deg_dinv, x, s1, N);
    gcn_scatter1<<<nblk_e, TB, 0, stream>>>(edges, deg_dinv, x, s1, E);

    const int n_tiles = (N + 15) / 16;               // one wave32 per tile
    const int waves_per_block = TB / 32;
    const int nblk_t = (n_tiles + waves_per_block - 1) / waves_per_block;
    gcn_node_wmma<<<nblk_t, TB, 0, stream>>>(s1, deg_dinv, W1, b1, W2,
                                             h2, h2acc, N, n_tiles);

    gcn_scatter2<<<nblk_e, TB, 0, stream>>>(edges, deg_dinv, h2, h2acc, E);
    gcn_final<<<nblk_n, TB, 0, stream>>>(h2acc, b2, out, N);
}